// MedianBlur_66932770340869
// MI455X (gfx1250) — compile-verified
//
#include <hip/hip_runtime.h>

// ---------------------------------------------------------------------------
// 3x3 median blur, fp32, zero padding, exact median-of-9 (19-op network):
//   sort each horizontal triple, then
//   median9 = med3( max(lo0,lo1,lo2), med3(mi0,mi1,mi2), min(hi0,hi1,hi2) )
//
// v3: branch-free halo. Each thread owns an aligned 4-wide x chunk and an
// 8-row strip; per row: ONE global_load_b128 + two UNCONDITIONAL clamped
// scalar halo loads (zero-pad applied via v_cndmask, mask is loop-invariant)
// + ONE global_store_b128. No divergent exec regions in the hot loop.
// Sorted row-triples roll through registers so each input row is loaded and
// sorted exactly once per thread.
//
// Roofline: 48 MiB in + 48 MiB out -> ~4.3 us floor at 23.3 TB/s; the whole
// input fits in the 192 MB L2 so strip-halo re-reads never touch HBM.
// ---------------------------------------------------------------------------

constexpr int H = 512;
constexpr int W = 512;

static __device__ __forceinline__ void sort3(float& a, float& b, float& c) {
    float t;
    t = fminf(a, b); b = fmaxf(a, b); a = t;   // a<=b
    t = fminf(b, c); c = fmaxf(b, c); b = t;   // c = max
    t = fminf(a, b); b = fmaxf(a, b); a = t;   // a = min, b = mid
}

static __device__ __forceinline__ float med3(float a, float b, float c) {
    return fmaxf(fminf(a, b), fminf(fmaxf(a, b), c));
}

template <int ROWS>
__global__ __launch_bounds__(128)
void MedianBlur_66932770340869_kernel(const float* __restrict__ in,
                                      float* __restrict__ out) {
    const int x  = (int)threadIdx.x * 4;          // aligned 4-wide chunk
    const int y0 = (int)blockIdx.y * ROWS;
    const size_t plane = (size_t)blockIdx.z * (size_t)(H * W);
    const float* __restrict__ p = in + plane;
    float* __restrict__ q = out + plane;

    // Loop-invariant halo addressing: clamped offsets + zero-select masks.
    const int  xl       = (x > 0) ? (x - 1) : 0;          // clamped left addr
    const int  xr       = (x + 4 < W) ? (x + 4) : (W - 1);// clamped right addr
    const bool has_left  = (x > 0);
    const bool has_right = (x + 4 < W);

    // Load v[0..5] = row y at x-1 .. x+4 with zero padding (branch-free).
    auto loadRow = [&](int y, float v[6]) {
        if ((unsigned)y >= (unsigned)H) {   // uniform (scalar) branch
#pragma unroll
            for (int j = 0; j < 6; ++j) v[j] = 0.f;
            return;
        }
        const float* __restrict__ r = p + (size_t)y * W;
        const float4 m = *reinterpret_cast<const float4*>(r + x);
        v[1] = m.x; v[2] = m.y; v[3] = m.z; v[4] = m.w;
        const float l = r[xl];              // unconditional, clamped
        const float rr = r[xr];             // unconditional, clamped
        v[0] = has_left  ? l  : 0.f;        // v_cndmask, loop-invariant mask
        v[5] = has_right ? rr : 0.f;
    };

    // Rolling window state: sorted triples for the top two rows, 4 lanes each.
    float lo0[4], mi0[4], hi0[4], lo1[4], mi1[4], hi1[4];
    {
        float v[6];
        loadRow(y0 - 1, v);
#pragma unroll
        for (int j = 0; j < 4; ++j) {
            float a = v[j], b = v[j + 1], c = v[j + 2];
            sort3(a, b, c);
            lo0[j] = a; mi0[j] = b; hi0[j] = c;
        }
    }
    {
        float v[6];
        loadRow(y0, v);
#pragma unroll
        for (int j = 0; j < 4; ++j) {
            float a = v[j], b = v[j + 1], c = v[j + 2];
            sort3(a, b, c);
            lo1[j] = a; mi1[j] = b; hi1[j] = c;
        }
    }

#pragma unroll
    for (int i = 0; i < ROWS; ++i) {
        const int y = y0 + i;

        // Pull the row two window-steps ahead toward the WGP while the
        // min/max network runs (lowers to global_prefetch_b8 on gfx1250).
        if (y + 3 < H) {
            __builtin_prefetch(p + (size_t)(y + 3) * W + x, 0, 1);
        }

        float v[6];
        loadRow(y + 1, v);

        float o[4];
#pragma unroll
        for (int j = 0; j < 4; ++j) {
            float a = v[j], b = v[j + 1], c = v[j + 2];
            sort3(a, b, c);                               // new bottom row
            const float mn = fmaxf(fmaxf(lo0[j], lo1[j]), a);  // max of minima
            const float md = med3(mi0[j], mi1[j], b);          // med of medians
            const float mx = fminf(fminf(hi0[j], hi1[j]), c);  // min of maxima
            o[j] = med3(mn, md, mx);
            // Roll the window for this lane.
            lo0[j] = lo1[j]; mi0[j] = mi1[j]; hi0[j] = hi1[j];
            lo1[j] = a;      mi1[j] = b;      hi1[j] = c;
        }

        *reinterpret_cast<float4*>(q + (size_t)y * W + x) =
            *reinterpret_cast<const float4*>(o);
    }
}

extern "C" void kernel_launch(void* const* d_in, const int* in_sizes, int n_in,
                              void* d_out, int out_size, void* d_ws, size_t ws_size,
                              hipStream_t stream) {
    (void)n_in; (void)out_size; (void)d_ws; (void)ws_size;

    const float* x = (const float*)d_in[0];
    float* y = (float*)d_out;

    constexpr int ROWS = 8;                       // output rows per strip
    const int planes = in_sizes[0] / (H * W);     // B*C = 48

    dim3 block(W / 4, 1, 1);                      // 128 threads = 4 wave32s
    dim3 grid(1,
              (H + ROWS - 1) / ROWS,              // 64
              planes);                            // 48  -> 3072 blocks

    MedianBlur_66932770340869_kernel<ROWS><<<grid, block, 0, stream>>>(x, y);
}